// Spacial3D_80607946211880
// MI455X (gfx1250) — compile-verified
//
#include <hip/hip_runtime.h>

// ---------------------------------------------------------------------------
// Swin-style window attention (B=2,T=8,W=64,N=64,C=96,H=8, D=W*C/H=768)
// Pipeline: qkv GEMM -> windowed attention (QK^T, +mask, softmax, PV) -> proj.
// All matrix math on V_WMMA_F32_16X16X4_F32 (fp32-exact vs the reference).
// ---------------------------------------------------------------------------

typedef __attribute__((ext_vector_type(2))) float v2f;
typedef __attribute__((ext_vector_type(8))) float v8f;

#define BTOT 16        // B*T
#define WN   64        // windows per (b,t)
#define NN   64        // tokens per window
#define CC   96        // channels
#define HH   8         // heads
#define DD   768       // effective head dim = W*C/H
#define C3   288       // 3*C
#define ROW3 18432     // per-window row of Y = N*3C = 3*H*D
#define YBT  1179648   // per-(bt) Y floats = 64*18432
#define HD   6144      // H*D
#define ZBT  393216    // per-(bt) Z floats = 64*6144
#define ATTN_SCALE 0.28867513459481287f   // (C/H)^-0.5 = 12^-0.5

static __device__ __forceinline__ v8f wmma4(v2f a, v2f b, v8f c) {
  // D(16x16,f32) = A(16x4,f32) * B(4x16,f32) + C
  return __builtin_amdgcn_wmma_f32_16x16x4_f32(
      /*neg_a=*/false, a, /*neg_b=*/false, b,
      /*c_mod=*/(short)0, c, /*reuse_a=*/false, /*reuse_b=*/false);
}

// ---------------------------------------------------------------------------
// Stage 1: Y[g][n][k] = sum_c X[g][n][c] * Wqkv[k][c]
//   g = bt*64 + w  (1024 groups), tile GEMM 64x96 @ 96x288
// Y is stored so that Q/K/V for (bt,h) are flat slices (see analysis).
// ---------------------------------------------------------------------------
__global__ __launch_bounds__(256) void qkv_gemm_kernel(
    const float* __restrict__ x, const float* __restrict__ wqkv,
    float* __restrict__ Y) {
  __shared__ __align__(16) float sX[64 * 100];   // padded rows: bank-conflict-free
  const int tid = threadIdx.x;
  const int lane = tid & 31, wave = tid >> 5;
  const int g = blockIdx.x;
  const float* xt = x + (size_t)g * (NN * CC);       // contiguous 64x96 tile
  float* yt = Y + (size_t)(g / WN) * YBT + (size_t)(g % WN) * ROW3;

  for (int i = tid; i < 64 * 24; i += 256) {          // 1536 float4s
    int r = i / 24, c4 = (i % 24) * 4;
    *reinterpret_cast<float4*>(&sX[r * 100 + c4]) =
        *reinterpret_cast<const float4*>(xt + r * CC + c4);
  }
  __syncthreads();

  const int m0 = lane & 15;
  const int ksel = (lane >> 4) * 2;
  for (int tt = wave; tt < 72; tt += 8) {             // 4 M-tiles x 18 N-tiles
    const int mt = tt & 3, nt = tt >> 2;
    const int ncol = nt * 16 + m0;
    const float* bp = wqkv + ncol * CC;               // B[k][n] = Wqkv[n][k]
    const float* ap = &sX[(mt * 16 + m0) * 100];
    v8f acc = {};
#pragma unroll
    for (int kk = 0; kk < 24; ++kk) {                 // K=96, 4 per WMMA
      const int kb = kk * 4 + ksel;
      v2f a = *reinterpret_cast<const v2f*>(ap + kb);
      v2f b = *reinterpret_cast<const v2f*>(bp + kb);
      acc = wmma4(a, b, acc);
    }
    float* yo = yt + ncol;
#pragma unroll
    for (int v = 0; v < 8; ++v) {
      const int m = mt * 16 + v + ((lane >> 4) << 3); // C/D layout
      yo[m * C3] = acc[v];
    }
  }
}

// ---------------------------------------------------------------------------
// Stage 2: per (bt,h): S = scale*Q*K^T + mask[bt%8]; P = softmax(S); O = P*V.
// Q[n][d] at Y + bt*YBT + n*ROW3 + h*DD; K at +HD; V at +2*HD. 128 blocks.
// ---------------------------------------------------------------------------
__global__ __launch_bounds__(256) void win_attn_kernel(
    const float* __restrict__ Y, const float* __restrict__ mask,
    float* __restrict__ Z) {
  __shared__ __align__(16) float sA[64 * 68];   // Q chunk (padded stride 68)
  __shared__ __align__(16) float sB[64 * 68];   // K then V chunk
  __shared__ __align__(16) float sS[64 * 68];   // scores / probabilities
  const int tid = threadIdx.x, lane = tid & 31, wave = tid >> 5;
  const int bt = blockIdx.x >> 3, h = blockIdx.x & 7;
  const float* qb = Y + (size_t)bt * YBT + h * DD;
  const float* kbp = qb + HD;
  const float* vbp = qb + 2 * HD;

  const int m0 = lane & 15;
  const int ksel = (lane >> 4) * 2;
  const int t0 = wave * 2;                      // 2 of 16 S-tiles per wave

  v8f accS[2];
  accS[0] = (v8f){};
  accS[1] = (v8f){};

  // ---- S = Q K^T over d in 12 chunks of 64 ----
  for (int dc = 0; dc < 12; ++dc) {
    for (int i = tid; i < 64 * 16; i += 256) {  // 64 rows x 16 float4
      const int r = i / 16, c4 = (i % 16) * 4;
      *reinterpret_cast<float4*>(&sA[r * 68 + c4]) =
          *reinterpret_cast<const float4*>(qb + r * ROW3 + dc * 64 + c4);
      *reinterpret_cast<float4*>(&sB[r * 68 + c4]) =
          *reinterpret_cast<const float4*>(kbp + r * ROW3 + dc * 64 + c4);
    }
    __syncthreads();
#pragma unroll
    for (int j = 0; j < 2; ++j) {
      const int tt = t0 + j, mt = tt >> 2, nt = tt & 3;
      const float* ap = &sA[(mt * 16 + m0) * 68];
      const float* bp = &sB[(nt * 16 + m0) * 68];   // B[k][n] = K[n][k]
#pragma unroll
      for (int kk = 0; kk < 16; ++kk) {
        const int kb = kk * 4 + ksel;
        v2f a = *reinterpret_cast<const v2f*>(ap + kb);
        v2f b = *reinterpret_cast<const v2f*>(bp + kb);
        accS[j] = wmma4(a, b, accS[j]);
      }
    }
    __syncthreads();
  }

  // ---- scale + additive mask, spill scores to LDS ----
  const float* mk = mask + (size_t)(bt & 7) * (NN * NN);
#pragma unroll
  for (int j = 0; j < 2; ++j) {
    const int tt = t0 + j, mt = tt >> 2, nt = tt & 3;
    const int ncol = nt * 16 + m0;
#pragma unroll
    for (int v = 0; v < 8; ++v) {
      const int m = mt * 16 + v + ((lane >> 4) << 3);
      sS[m * 68 + ncol] = accS[j][v] * ATTN_SCALE + mk[m * NN + ncol];
    }
  }
  __syncthreads();

  // ---- row softmax: each wave32 owns 8 rows, 2 cols per lane ----
  for (int r = wave * 8; r < wave * 8 + 8; ++r) {
    float x0 = sS[r * 68 + lane];
    float x1 = sS[r * 68 + lane + 32];
    float mx = fmaxf(x0, x1);
#pragma unroll
    for (int off = 16; off >= 1; off >>= 1)
      mx = fmaxf(mx, __shfl_xor(mx, off, 32));
    const float e0 = __expf(x0 - mx), e1 = __expf(x1 - mx);
    float sm = e0 + e1;
#pragma unroll
    for (int off = 16; off >= 1; off >>= 1)
      sm += __shfl_xor(sm, off, 32);
    const float inv = 1.0f / sm;
    sS[r * 68 + lane] = e0 * inv;
    sS[r * 68 + lane + 32] = e1 * inv;
  }
  __syncthreads();

  // ---- O = P V, chunked over d; store directly into Z ----
  float* zb = Z + (size_t)bt * ZBT + h * DD;    // element (n,d) at n*HD + d
  for (int dc = 0; dc < 12; ++dc) {
    for (int i = tid; i < 64 * 16; i += 256) {
      const int r = i / 16, c4 = (i % 16) * 4;
      *reinterpret_cast<float4*>(&sB[r * 68 + c4]) =
          *reinterpret_cast<const float4*>(vbp + r * ROW3 + dc * 64 + c4);
    }
    __syncthreads();
#pragma unroll
    for (int j = 0; j < 2; ++j) {
      const int tt = t0 + j, mt = tt >> 2, dt = tt & 3;
      const float* ap = &sS[(mt * 16 + m0) * 68];
      v8f acc = {};
#pragma unroll
      for (int kk = 0; kk < 16; ++kk) {
        const int kb = kk * 4 + ksel;
        v2f a = *reinterpret_cast<const v2f*>(ap + kb);
        v2f b;
        b.x = sB[kb * 68 + dt * 16 + m0];       // B[k][d] = V[k][d]
        b.y = sB[(kb + 1) * 68 + dt * 16 + m0];
        acc = wmma4(a, b, acc);
      }
      const int dcol = dc * 64 + dt * 16 + m0;
#pragma unroll
      for (int v = 0; v < 8; ++v) {
        const int m = mt * 16 + v + ((lane >> 4) << 3);
        zb[m * HD + dcol] = acc[v];
      }
    }
    __syncthreads();
  }
}

// ---------------------------------------------------------------------------
// Stage 3: out[g][n][o] = sum_c Z[g][n][c] * Wproj[o][c] + bproj[o]
//   g = bt*64 + w (Z tile is contiguous 64x96), 1024 groups.
// ---------------------------------------------------------------------------
__global__ __launch_bounds__(256) void proj_gemm_kernel(
    const float* __restrict__ Z, const float* __restrict__ wproj,
    const float* __restrict__ bproj, float* __restrict__ out) {
  __shared__ __align__(16) float sZ[64 * 100];
  const int tid = threadIdx.x, lane = tid & 31, wave = tid >> 5;
  const int g = blockIdx.x;
  const float* zt = Z + (size_t)g * (NN * CC);
  float* ot = out + (size_t)g * (NN * CC);

  for (int i = tid; i < 64 * 24; i += 256) {
    const int r = i / 24, c4 = (i % 24) * 4;
    *reinterpret_cast<float4*>(&sZ[r * 100 + c4]) =
        *reinterpret_cast<const float4*>(zt + r * CC + c4);
  }
  __syncthreads();

  const int m0 = lane & 15;
  const int ksel = (lane >> 4) * 2;
  for (int tt = wave; tt < 24; tt += 8) {       // 4 M-tiles x 6 N-tiles
    const int mt = tt & 3, nt = tt >> 2;
    const int ocol = nt * 16 + m0;
    const float* bp = wproj + ocol * CC;        // B[c][o] = Wproj[o][c]
    const float* ap = &sZ[(mt * 16 + m0) * 100];
    v8f acc = {};
#pragma unroll
    for (int kk = 0; kk < 24; ++kk) {
      const int kb = kk * 4 + ksel;
      v2f a = *reinterpret_cast<const v2f*>(ap + kb);
      v2f b = *reinterpret_cast<const v2f*>(bp + kb);
      acc = wmma4(a, b, acc);
    }
    const float bias = bproj[ocol];
#pragma unroll
    for (int v = 0; v < 8; ++v) {
      const int m = mt * 16 + v + ((lane >> 4) << 3);
      ot[m * CC + ocol] = acc[v] + bias;
    }
  }
}

// ---------------------------------------------------------------------------
extern "C" void kernel_launch(void* const* d_in, const int* in_sizes, int n_in,
                              void* d_out, int out_size, void* d_ws, size_t ws_size,
                              hipStream_t stream) {
  (void)in_sizes; (void)n_in; (void)out_size; (void)ws_size;
  const float* x     = (const float*)d_in[0];   // [2,8,64,64,96]
  const float* mask  = (const float*)d_in[1];   // [8,64,64]
  const float* wqkv  = (const float*)d_in[2];   // [288,96]
  const float* wproj = (const float*)d_in[3];   // [96,96]
  const float* bproj = (const float*)d_in[4];   // [96]
  float* out = (float*)d_out;                   // [16,64,64,96]

  float* Y = (float*)d_ws;                      // 16*1179648 f32 = 75.5 MB
  float* Z = Y + (size_t)BTOT * YBT;            // 16*393216  f32 = 25.2 MB

  qkv_gemm_kernel<<<dim3(BTOT * WN), dim3(256), 0, stream>>>(x, wqkv, Y);
  win_attn_kernel<<<dim3(BTOT * HH), dim3(256), 0, stream>>>(Y, mask, Z);
  proj_gemm_kernel<<<dim3(BTOT * WN), dim3(256), 0, stream>>>(Z, wproj, bproj, out);
}